// FixedRateDelta_34583076667904
// MI455X (gfx1250) — compile-verified
//
#include <hip/hip_runtime.h>
#include <hip/hip_bf16.h>

typedef float v2f __attribute__((ext_vector_type(2)));
typedef float v8f __attribute__((ext_vector_type(8)));

#define H     64
#define LSEQ  1024
#define BATCH 64
#define NVOC  64

__device__ __forceinline__ v8f wmma_f32(v2f a, v2f b, v8f c) {
  // D = A(16x4, f32) * B(4x16, f32) + C(16x16, f32), exact f32 MACs
  return __builtin_amdgcn_wmma_f32_16x16x4_f32(
      /*neg_a=*/false, a, /*neg_b=*/false, b,
      /*c_mod=*/(short)0, c, /*reuse_a=*/false, /*reuse_b=*/false);
}

__device__ __forceinline__ v8f splat8(float x) {
  v8f t;
#pragma unroll
  for (int e = 0; e < 8; ++e) t[e] = x;
  return t;
}

// ---------------------------------------------------------------------------
// Kernel 1: h = embed_W[seq]; ff = relu(h@W1+b1)@W2+b2; x = h+ff; hn = LN(x)
// 256 threads = 8 waves, each wave owns a 16-token tile (128 tokens / block).
// WMMA f32 16x16x4: A-frag lane(l): row m=l%16, a={A[m][k0],A[m][k0+1]},
// k0 = 4*kk + 2*(l/16).  B-frag: col n=l%16, b={B[k0][n],B[k0+1][n]}.
// C/D: reg r -> (m = r + 8*(l/16), n = l%16).
// ---------------------------------------------------------------------------
__global__ __launch_bounds__(256)
void k_embed_ff_ln(const int* __restrict__ seq, const float* __restrict__ embW,
                   const float* __restrict__ W1, const float* __restrict__ b1,
                   const float* __restrict__ W2, const float* __restrict__ b2,
                   const float* __restrict__ lng, const float* __restrict__ lnb,
                   float* __restrict__ hn)
{
  __shared__ float sFF[8][16 * 128];   // per-wave relu(h@W1+b1) tile, 64 KB

  const int wave = threadIdx.x >> 5;
  const int lane = threadIdx.x & 31;
  const int half = lane >> 4;
  const int lm   = lane & 15;
  const int tokbase = blockIdx.x * 128 + wave * 16;

  const int sA = seq[tokbase + lm];    // embed row for this lane's A-row

  // ---- GEMM1: [16x64] @ [64x128] + b1, ReLU ----
  v8f acc[8];
#pragma unroll
  for (int n = 0; n < 8; ++n) acc[n] = splat8(b1[n * 16 + lm]);

#pragma unroll
  for (int kk = 0; kk < 16; ++kk) {
    const int k0 = kk * 4 + half * 2;
    v2f a;
    a.x = embW[sA * H + k0];
    a.y = embW[sA * H + k0 + 1];
#pragma unroll
    for (int n = 0; n < 8; ++n) {
      v2f bf;
      bf.x = W1[k0 * 128 + n * 16 + lm];
      bf.y = W1[(k0 + 1) * 128 + n * 16 + lm];
      acc[n] = wmma_f32(a, bf, acc[n]);
    }
  }
#pragma unroll
  for (int n = 0; n < 8; ++n) {
#pragma unroll
    for (int r = 0; r < 8; ++r) {
      float v = acc[n][r];
      sFF[wave][(r + 8 * half) * 128 + n * 16 + lm] = v > 0.f ? v : 0.f;
    }
  }
  __syncthreads();

  // ---- GEMM2: [16x128] @ [128x64] + b2 ----
  v8f acc2[4];
#pragma unroll
  for (int n = 0; n < 4; ++n) acc2[n] = splat8(b2[n * 16 + lm]);

#pragma unroll
  for (int kk = 0; kk < 32; ++kk) {
    const int k0 = kk * 4 + half * 2;
    v2f a;
    a.x = sFF[wave][lm * 128 + k0];
    a.y = sFF[wave][lm * 128 + k0 + 1];
#pragma unroll
    for (int n = 0; n < 4; ++n) {
      v2f bf;
      bf.x = W2[k0 * H + n * 16 + lm];
      bf.y = W2[(k0 + 1) * H + n * 16 + lm];
      acc2[n] = wmma_f32(a, bf, acc2[n]);
    }
  }

  // ---- residual + LayerNorm (per-row stats over H=64) ----
  float gv[4], bv[4];
#pragma unroll
  for (int n = 0; n < 4; ++n) { gv[n] = lng[n * 16 + lm]; bv[n] = lnb[n * 16 + lm]; }

#pragma unroll
  for (int r = 0; r < 8; ++r) {
    const int m = r + 8 * half;
    const int s = seq[tokbase + m];
    float x[4], s1 = 0.f, s2 = 0.f;
#pragma unroll
    for (int n = 0; n < 4; ++n) {
      x[n] = acc2[n][r] + embW[s * H + n * 16 + lm];   // residual h + ff
      s1 += x[n];
      s2 += x[n] * x[n];
    }
    // reduce across the 16 lanes of this half (row m lives entirely there)
#pragma unroll
    for (int mask = 8; mask >= 1; mask >>= 1) {
      s1 += __shfl_xor(s1, mask, 32);
      s2 += __shfl_xor(s2, mask, 32);
    }
    const float mu   = s1 * (1.f / 64.f);
    const float var  = s2 * (1.f / 64.f) - mu * mu;
    const float rstd = rsqrtf(var + 1e-5f);
#pragma unroll
    for (int n = 0; n < 4; ++n)
      hn[(tokbase + m) * H + n * 16 + lm] = (x[n] - mu) * rstd * gv[n] + bv[n];
  }
}

// ---------------------------------------------------------------------------
// Kernel 2: k = hn@kp_W (then L2-normalize rows, eps=1e-12), v = hn@vp_W
// Same WMMA tiling as kernel 1. Computes all L rows (last row unused by scan).
// ---------------------------------------------------------------------------
__global__ __launch_bounds__(256)
void k_kv(const float* __restrict__ hn, const float* __restrict__ kpW,
          const float* __restrict__ vpW, float* __restrict__ knO,
          float* __restrict__ vO)
{
  const int wave = threadIdx.x >> 5;
  const int lane = threadIdx.x & 31;
  const int half = lane >> 4;
  const int lm   = lane & 15;
  const int tokbase = blockIdx.x * 128 + wave * 16;

  v8f ak[4], av[4];
#pragma unroll
  for (int n = 0; n < 4; ++n) { ak[n] = splat8(0.f); av[n] = splat8(0.f); }

#pragma unroll
  for (int kk = 0; kk < 16; ++kk) {
    const int k0 = kk * 4 + half * 2;
    v2f a;
    a.x = hn[(tokbase + lm) * H + k0];
    a.y = hn[(tokbase + lm) * H + k0 + 1];
#pragma unroll
    for (int n = 0; n < 4; ++n) {
      v2f bk, bv2;
      bk.x  = kpW[k0 * H + n * 16 + lm];
      bk.y  = kpW[(k0 + 1) * H + n * 16 + lm];
      bv2.x = vpW[k0 * H + n * 16 + lm];
      bv2.y = vpW[(k0 + 1) * H + n * 16 + lm];
      ak[n] = wmma_f32(a, bk, ak[n]);
      av[n] = wmma_f32(a, bv2, av[n]);
    }
  }

#pragma unroll
  for (int r = 0; r < 8; ++r) {
    const int m = r + 8 * half;
    float s2 = 0.f;
#pragma unroll
    for (int n = 0; n < 4; ++n) { float kv = ak[n][r]; s2 += kv * kv; }
#pragma unroll
    for (int mask = 8; mask >= 1; mask >>= 1) s2 += __shfl_xor(s2, mask, 32);
    const float inv = 1.f / fmaxf(sqrtf(s2), 1e-12f);
#pragma unroll
    for (int n = 0; n < 4; ++n) {
      knO[(tokbase + m) * H + n * 16 + lm] = ak[n][r] * inv;
      vO [(tokbase + m) * H + n * 16 + lm] = av[n][r];
    }
  }
}

// ---------------------------------------------------------------------------
// Kernel 3: sequential fast-weight scan, SINGLE WAVE per batch.
// Lane l owns M rows 2l and 2l+1 (128 VGPRs). Only kn needs LDS broadcast;
// v stays in registers (each row's v is consumed only by its owning lane).
// Gate reduction is pure wave32 shuffles; barriers are S_NOP (1-wave WG).
// Gate: err^2 > 0.16*||v||^2  (== err > 0.4*||v||, both non-negative).
// ---------------------------------------------------------------------------
__global__ __launch_bounds__(32)
void k_scan(const float* __restrict__ kn, const float* __restrict__ vv,
            float* __restrict__ Mout)
{
  const int b = blockIdx.x;
  const int l = threadIdx.x & 31;

  __shared__ v2f kb[2][32];            // double-buffered kn (64 f32 each)

  float M0[64], M1[64];
#pragma unroll
  for (int j = 0; j < 64; ++j) { M0[j] = 0.f; M1[j] = 0.f; }

  const int base = b * LSEQ * H;

  // preload t=0 (kn broadcast buffer + private v)
  v2f k0v = *(const v2f*)(kn + base + 2 * l);
  v2f vcur = *(const v2f*)(vv + base + 2 * l);
  kb[0][l] = k0v;
  __syncthreads();

  for (int t = 0; t < LSEQ - 1; ++t) {
    const int cur = t & 1, nxt = cur ^ 1;

    // unconditional prefetch of step t+1 (row t+1 <= L-1 is valid memory)
    const v2f knext = *(const v2f*)(kn + base + (t + 1) * H + 2 * l);
    const v2f vnext = *(const v2f*)(vv + base + (t + 1) * H + 2 * l);

    // pull kn_t into registers once; reused by dot and rank-1 update
    float kr[64];
#pragma unroll
    for (int j = 0; j < 32; ++j) {
      v2f p = kb[cur][j];
      kr[2 * j]     = p.x;
      kr[2 * j + 1] = p.y;
    }

    float vp0 = 0.f, vp1 = 0.f;
#pragma unroll
    for (int j = 0; j < 64; ++j) {
      vp0 = fmaf(M0[j], kr[j], vp0);
      vp1 = fmaf(M1[j], kr[j], vp1);
    }

    const float d0 = vcur.x - vp0;
    const float d1 = vcur.y - vp1;
    float e2  = d0 * d0 + d1 * d1;
    float vn2 = vcur.x * vcur.x + vcur.y * vcur.y;
#pragma unroll
    for (int mask = 16; mask >= 1; mask >>= 1) {
      e2  += __shfl_xor(e2,  mask, 32);
      vn2 += __shfl_xor(vn2, mask, 32);
    }
    const float g   = (e2 > 0.16f * vn2) ? 1.f : 0.f;
    const float gd0 = g * d0;
    const float gd1 = g * d1;
#pragma unroll
    for (int j = 0; j < 64; ++j) {
      M0[j] = fmaf(gd0, kr[j], M0[j]);
      M1[j] = fmaf(gd1, kr[j], M1[j]);
    }

    kb[nxt][l] = knext;                // stage next step's kn
    vcur = vnext;                      // next step's v stays in registers
    __syncthreads();                   // S_NOP + waitcnt (single-wave WG)
  }

#pragma unroll
  for (int j = 0; j < 64; ++j) {
    Mout[b * 4096 + (2 * l) * 64 + j]     = M0[j];
    Mout[b * 4096 + (2 * l + 1) * 64 + j] = M1[j];
  }
}

// ---------------------------------------------------------------------------
// Kernel 4: q = hn[:, -1] @ qp_W; r = (M @ q) @ rp_W + rp_b; out = r@out_W+out_b
// One block of 64 threads per batch; tiny, LDS-staged GEMVs.
// ---------------------------------------------------------------------------
__global__ __launch_bounds__(64)
void k_final(const float* __restrict__ hn, const float* __restrict__ Mw,
             const float* __restrict__ qpW, const float* __restrict__ rpW,
             const float* __restrict__ rpb, const float* __restrict__ outW,
             const float* __restrict__ outb, float* __restrict__ out)
{
  const int b = blockIdx.x, i = threadIdx.x;
  __shared__ float sh[H], sq[H], smq[H], sr[H];

  sh[i] = hn[(b * LSEQ + (LSEQ - 1)) * H + i];
  __syncthreads();

  float q = 0.f;
#pragma unroll
  for (int j = 0; j < 64; ++j) q = fmaf(sh[j], qpW[j * H + i], q);
  sq[i] = q; __syncthreads();

  float mq = 0.f;
#pragma unroll
  for (int j = 0; j < 64; ++j) mq = fmaf(Mw[b * 4096 + i * 64 + j], sq[j], mq);
  smq[i] = mq; __syncthreads();

  float rv = rpb[i];
#pragma unroll
  for (int j = 0; j < 64; ++j) rv = fmaf(smq[j], rpW[j * H + i], rv);
  sr[i] = rv; __syncthreads();

  float ov = outb[i];
#pragma unroll
  for (int j = 0; j < 64; ++j) ov = fmaf(sr[j], outW[j * NVOC + i], ov);
  out[b * NVOC + i] = ov;
}

// ---------------------------------------------------------------------------
extern "C" void kernel_launch(void* const* d_in, const int* in_sizes, int n_in,
                              void* d_out, int out_size, void* d_ws, size_t ws_size,
                              hipStream_t stream) {
  const int*   seq  = (const int*)  d_in[0];
  const float* embW = (const float*)d_in[1];
  const float* W1   = (const float*)d_in[2];
  const float* b1   = (const float*)d_in[3];
  const float* W2   = (const float*)d_in[4];
  const float* b2   = (const float*)d_in[5];
  const float* lng  = (const float*)d_in[6];
  const float* lnb  = (const float*)d_in[7];
  const float* kpW  = (const float*)d_in[8];
  const float* vpW  = (const float*)d_in[9];
  const float* qpW  = (const float*)d_in[10];
  const float* rpW  = (const float*)d_in[11];
  const float* rpb  = (const float*)d_in[12];
  const float* outW = (const float*)d_in[13];
  const float* outb = (const float*)d_in[14];

  float* ws  = (float*)d_ws;
  float* hn  = ws;                                       // 64*1024*64 f32
  float* knA = hn  + (size_t)BATCH * LSEQ * H;           // 64*1024*64 f32
  float* vA  = knA + (size_t)BATCH * LSEQ * H;           // 64*1024*64 f32
  float* Mw  = vA  + (size_t)BATCH * LSEQ * H;           // 64*64*64 f32

  k_embed_ff_ln<<<512, 256, 0, stream>>>(seq, embW, W1, b1, W2, b2, lng, lnb, hn);
  k_kv        <<<512, 256, 0, stream>>>(hn, kpW, vpW, knA, vA);
  k_scan      <<<BATCH, 32, 0, stream>>>(knA, vA, Mw);
  k_final     <<<BATCH, 64, 0, stream>>>(hn, Mw, qpW, rpW, rpb, outW, outb,
                                         (float*)d_out);
}